// AURGCN_20143396618489
// MI455X (gfx1250) — compile-verified
//
#include <hip/hip_runtime.h>
#include <hip/hip_bf16.h>

#define N_NODES   50000
#define N_EDGES   800000
#define EMB       64
#define HEADS     4
#define LAYERS    2
#define N_RELS    200
#define NUM_GRAPHS 64
#define NEG_SLOPE 0.2f

typedef __attribute__((ext_vector_type(16))) _Float16 v16h;
typedef __attribute__((ext_vector_type(8)))  float    v8f;
typedef __attribute__((ext_vector_type(2)))  float    v2f;

#define GEMM_WAVES 8
#define USE_ASYNC_LDS 1

// ---------------------------------------------------------------------------
// C[M,N] = A[M,64] * B[64,N], all f32 row-major (N <= 256, multiple of 16).
// B is staged into LDS once per workgroup with gfx1250 async global->LDS
// copies (ASYNCcnt), then each wave32 computes one 16-row tile of C with
// V_WMMA_F32_16X16X4_F32: A row fragments live in 32 VGPRs for the whole
// kernel, B fragments are pre-read from LDS per 16-column tile so the 16
// WMMAs chain back-to-back on the accumulator only.
// ---------------------------------------------------------------------------
__global__ __launch_bounds__(32 * GEMM_WAVES)
void gemm64_wmma(const float* __restrict__ A, const float* __restrict__ B,
                 float* __restrict__ C, int M, int N) {
  __shared__ float smemB[64 * 256];   // max 64KB; single shared array
  const int tid = threadIdx.x;

  // ---- stage B[64][N] into LDS ----
  const int total4 = (64 * N) >> 2;          // number of float4 chunks
  for (int idx = tid; idx < total4; idx += 32 * GEMM_WAVES) {
#if USE_ASYNC_LDS
    // Generic-address low 32 bits == wave-relative LDS byte offset.
    // ptrtoint captures smemB so the "memory" clobber is known to cover it.
    unsigned loff = (unsigned)(size_t)(&smemB[idx * 4]);
    const float* g = B + (size_t)idx * 4;
    asm volatile("global_load_async_to_lds_b128 %0, %1, off"
                 :: "v"(loff), "v"(g), "v"(smemB) : "memory");
#else
    ((float4*)smemB)[idx] = ((const float4*)B)[idx];
#endif
  }
#if USE_ASYNC_LDS
  asm volatile("s_wait_asynccnt 0x0" ::: "memory");
#endif
  __syncthreads();

  const int lane = tid & 31;
  const int wave = tid >> 5;
  const int mt   = blockIdx.x * GEMM_WAVES + wave;
  if (mt * 16 < M) {                         // wave-uniform: EXEC stays all-1s
    const int half = lane >> 4;              // 0: lanes 0-15, 1: lanes 16-31
    const int l15  = lane & 15;

    // A 16x4 fragment layout: lanes0-15 hold K={0,1}, lanes16-31 K={2,3}
    float areg[32];
    const float* Ar = A + (size_t)(mt * 16 + l15) * EMB + half * 2;
#pragma unroll
    for (int t = 0; t < 16; ++t) {
      areg[2 * t]     = Ar[4 * t];
      areg[2 * t + 1] = Ar[4 * t + 1];
    }

    for (int n0 = 0; n0 < N; n0 += 16) {
      // pre-read all 16 B fragments for this column tile from LDS
      float bx[16], by[16];
      const float* Bp = smemB + (half * 2) * N + n0 + l15;
#pragma unroll
      for (int t = 0; t < 16; ++t) {
        bx[t] = Bp[(4 * t) * N];
        by[t] = Bp[(4 * t + 1) * N];
      }
      v8f c = {};
#pragma unroll
      for (int t = 0; t < 16; ++t) {
        v2f a, b;
        a.x = areg[2 * t];
        a.y = areg[2 * t + 1];
        b.x = bx[t];
        b.y = by[t];
        c = __builtin_amdgcn_wmma_f32_16x16x4_f32(false, a, false, b, (short)0, c,
                                                  false, false);
      }
      float* Cp = C + (size_t)(mt * 16 + half * 8) * N + n0 + l15;
#pragma unroll
      for (int r = 0; r < 8; ++r) Cp[(size_t)r * N] = c[r];
    }
  }
}

// ---------------------------------------------------------------------------
// small helper kernels
// ---------------------------------------------------------------------------
__global__ void k_zero_f32(float* p, int n) {
  int i = blockIdx.x * blockDim.x + threadIdx.x;
  if (i < n) p[i] = 0.f;
}
__global__ void k_zero_i32(int* p, int n) {
  int i = blockIdx.x * blockDim.x + threadIdx.x;
  if (i < n) p[i] = 0;
}
__global__ void k_fill_neginf(float* p, int n) {
  int i = blockIdx.x * blockDim.x + threadIdx.x;
  if (i < n) p[i] = -__builtin_inff();
}

__global__ void k_indeg(const int* __restrict__ dst, int* __restrict__ indeg) {
  int e = blockIdx.x * blockDim.x + threadIdx.x;
  if (e < N_EDGES) atomicAdd(&indeg[dst[e]], 1);
}

__global__ void k_gather(const int* __restrict__ node_ids,
                         const float* __restrict__ code_emb,
                         float* __restrict__ h) {
  int i = blockIdx.x * blockDim.x + threadIdx.x;
  if (i >= N_NODES * EMB) return;
  int n = i >> 6, d = i & 63;
  h[i] = code_emb[(size_t)node_ids[n] * EMB + d];
}

// Wpack[k][c]: c=0..3 -> attn_l head c, 4..7 -> attn_r, 8..11 -> attn_s, folded
// through fc_w so el/er/es become [N,64]x[64,4] products.
__global__ void k_prep_w(const float* __restrict__ fc_w,
                         const float* __restrict__ attn_l,
                         const float* __restrict__ attn_r,
                         const float* __restrict__ attn_s, int layer,
                         float* __restrict__ Wpack) {
  int i = blockIdx.x * blockDim.x + threadIdx.x;
  if (i >= EMB * 12) return;
  int k = i / 12, c = i % 12, hh = c & 3;
  const float* attn = (c < 4) ? attn_l : (c < 8) ? attn_r : attn_s;
  const float* fw = fc_w + (size_t)layer * EMB * HEADS * EMB + (size_t)k * HEADS * EMB + hh * EMB;
  const float* av = attn + (size_t)layer * HEADS * EMB + hh * EMB;
  float s = 0.f;
  for (int d = 0; d < EMB; ++d) s += fw[d] * av[d];
  Wpack[k * 12 + c] = s;
}

// relmsg[r][:] = rel_emb[r] @ weight_neighbor[layer]
__global__ void k_relmsg(const float* __restrict__ rel_emb,
                         const float* __restrict__ w_nb, int layer,
                         float* __restrict__ relmsg) {
  int i = blockIdx.x * blockDim.x + threadIdx.x;
  if (i >= N_RELS * EMB) return;
  int r = i >> 6, d = i & 63;
  const float* wn = w_nb + (size_t)layer * EMB * EMB;
  float s = 0.f;
  for (int k = 0; k < EMB; ++k) s += rel_emb[r * EMB + k] * wn[k * EMB + d];
  relmsg[i] = s;
}

// scores[n][0:4]=el, [4:8]=er (from h), [8:12]=es (from state)
__global__ void k_scores(const float* __restrict__ h,
                         const float* __restrict__ state,
                         const float* __restrict__ Wpack,
                         float* __restrict__ scores) {
  int i = blockIdx.x * blockDim.x + threadIdx.x;
  if (i >= N_NODES * 12) return;
  int n = i / 12, c = i % 12;
  const float* X = (c < 8) ? (h + (size_t)n * EMB) : (state + (size_t)n * EMB);
  float s = 0.f;
  for (int k = 0; k < EMB; ++k) s += X[k] * Wpack[k * 12 + c];
  scores[i] = s;
}

__device__ __forceinline__ void atomicMaxF(float* addr, float v) {
  // IEEE ordering trick: signed-int max for v>=0, unsigned-int min for v<0
  if (v >= 0.f) atomicMax((int*)addr, __float_as_int(v));
  else          atomicMin((unsigned int*)addr, __float_as_uint(v));
}

__global__ void k_edge_max(const int* __restrict__ src, const int* __restrict__ dst,
                           const float* __restrict__ scores,
                           float* __restrict__ ebuf, float* __restrict__ m) {
  int i = blockIdx.x * blockDim.x + threadIdx.x;
  if (i >= N_EDGES * HEADS) return;
  int e = i >> 2, hd = i & 3;
  int s = src[e], dd = dst[e];
  float v = scores[s * 12 + 8 + hd] + scores[s * 12 + hd] + scores[dd * 12 + 4 + hd];
  v = (v > 0.f) ? v : NEG_SLOPE * v;   // leaky_relu
  ebuf[i] = v;
  atomicMaxF(&m[dd * HEADS + hd], v);
}

__global__ void k_edge_exp(const int* __restrict__ dst, const float* __restrict__ m,
                           float* __restrict__ ebuf, float* __restrict__ z) {
  int i = blockIdx.x * blockDim.x + threadIdx.x;
  if (i >= N_EDGES * HEADS) return;
  int e = i >> 2, hd = i & 3;
  int dd = dst[e];
  float ex = __expf(ebuf[i] - m[dd * HEADS + hd]);
  ebuf[i] = ex;
  atomicAdd(&z[dd * HEADS + hd], ex);
}

// msg[e][d] = sum_h (ee/z)*feat[src][h][d] + relmsg[rel][d]; scatter-add to agg[dst]
__global__ void k_edge_msg(const int* __restrict__ src, const int* __restrict__ dst,
                           const int* __restrict__ rel,
                           const float* __restrict__ ebuf, const float* __restrict__ z,
                           const float* __restrict__ feat,
                           const float* __restrict__ relmsg,
                           float* __restrict__ agg) {
  long i = (long)blockIdx.x * blockDim.x + threadIdx.x;
  if (i >= (long)N_EDGES * EMB) return;
  int e = (int)(i >> 6), d = (int)(i & 63);
  int s = src[e], dd = dst[e];
  float acc = relmsg[rel[e] * EMB + d];
#pragma unroll
  for (int hh = 0; hh < HEADS; ++hh) {
    float a = ebuf[e * HEADS + hh] / z[dd * HEADS + hh];
    acc += a * feat[(size_t)s * (HEADS * EMB) + hh * EMB + d];
  }
  atomicAdd(&agg[(size_t)dd * EMB + d], acc);
}

__global__ void k_combine(const float* __restrict__ agg,
                          const float* __restrict__ loopA,
                          const float* __restrict__ loopB,
                          const int* __restrict__ indeg, float* __restrict__ hout) {
  int i = blockIdx.x * blockDim.x + threadIdx.x;
  if (i >= N_NODES * EMB) return;
  int n = i >> 6;
  float v = agg[i] + (indeg[n] > 0 ? loopA[i] : loopB[i]);
  hout[i] = v > 0.f ? v : 0.f;   // relu
}

__global__ void k_readout(const int* __restrict__ graph_id,
                          const float* __restrict__ h,
                          const float* __restrict__ w, float* __restrict__ out) {
  int i = blockIdx.x * blockDim.x + threadIdx.x;
  if (i >= N_NODES * EMB) return;
  int n = i >> 6, d = i & 63;
  atomicAdd(&out[graph_id[n] * EMB + d], h[i] * w[n]);
}

// ---------------------------------------------------------------------------
extern "C" void kernel_launch(void* const* d_in, const int* in_sizes, int n_in,
                              void* d_out, int out_size, void* d_ws, size_t ws_size,
                              hipStream_t stream) {
  (void)in_sizes; (void)n_in; (void)out_size; (void)ws_size;
  const int*   node_ids = (const int*)d_in[0];
  const int*   src      = (const int*)d_in[1];
  const int*   dst      = (const int*)d_in[2];
  const int*   rel      = (const int*)d_in[3];
  const int*   graph_id = (const int*)d_in[4];
  const float* state    = (const float*)d_in[5];
  const float* rd_w     = (const float*)d_in[6];
  const float* code_emb = (const float*)d_in[7];
  const float* rel_emb  = (const float*)d_in[8];
  const float* fc_w     = (const float*)d_in[9];
  const float* attn_l   = (const float*)d_in[10];
  const float* attn_r   = (const float*)d_in[11];
  const float* attn_s   = (const float*)d_in[12];
  const float* w_nb     = (const float*)d_in[13];
  const float* loop_w   = (const float*)d_in[14];
  const float* evol_w   = (const float*)d_in[15];
  float* out = (float*)d_out;

  char* wsp = (char*)d_ws;
  auto alloc = [&](size_t bytes) -> char* {
    char* p = wsp;
    wsp += (bytes + 255) & ~(size_t)255;
    return p;
  };
  float* h0     = (float*)alloc(sizeof(float) * N_NODES * EMB);
  float* h1     = (float*)alloc(sizeof(float) * N_NODES * EMB);
  float* feat   = (float*)alloc(sizeof(float) * (size_t)N_NODES * HEADS * EMB);
  float* scores = (float*)alloc(sizeof(float) * N_NODES * 12);
  float* mbuf   = (float*)alloc(sizeof(float) * N_NODES * HEADS);
  float* zbuf   = (float*)alloc(sizeof(float) * N_NODES * HEADS);
  float* ebuf   = (float*)alloc(sizeof(float) * (size_t)N_EDGES * HEADS);
  float* agg    = (float*)alloc(sizeof(float) * N_NODES * EMB);
  float* loopA  = (float*)alloc(sizeof(float) * N_NODES * EMB);
  float* loopB  = (float*)alloc(sizeof(float) * N_NODES * EMB);
  float* relmsg = (float*)alloc(sizeof(float) * N_RELS * EMB);
  float* Wpack  = (float*)alloc(sizeof(float) * EMB * 12);
  int*   indeg  = (int*)alloc(sizeof(int) * N_NODES);

  const int B = 256;
  auto G = [](long n, int b) { return (int)((n + b - 1) / b); };
  const int gemm_blocks = G(N_NODES / 16, GEMM_WAVES);

  k_zero_i32<<<G(N_NODES, B), B, 0, stream>>>(indeg, N_NODES);
  k_indeg<<<G(N_EDGES, B), B, 0, stream>>>(dst, indeg);
  k_gather<<<G((long)N_NODES * EMB, B), B, 0, stream>>>(node_ids, code_emb, h0);

  float* hin = h0;
  float* hout = h1;
  for (int l = 0; l < LAYERS; ++l) {
    k_prep_w<<<G(EMB * 12, B), B, 0, stream>>>(fc_w, attn_l, attn_r, attn_s, l, Wpack);
    k_relmsg<<<G(N_RELS * EMB, B), B, 0, stream>>>(rel_emb, w_nb, l, relmsg);
    k_fill_neginf<<<G(N_NODES * HEADS, B), B, 0, stream>>>(mbuf, N_NODES * HEADS);
    k_zero_f32<<<G(N_NODES * HEADS, B), B, 0, stream>>>(zbuf, N_NODES * HEADS);
    k_zero_f32<<<G((long)N_NODES * EMB, B), B, 0, stream>>>(agg, N_NODES * EMB);

    gemm64_wmma<<<gemm_blocks, 32 * GEMM_WAVES, 0, stream>>>(
        hin, fc_w + (size_t)l * EMB * HEADS * EMB, feat, N_NODES, HEADS * EMB);
    k_scores<<<G((long)N_NODES * 12, B), B, 0, stream>>>(hin, state, Wpack, scores);
    k_edge_max<<<G((long)N_EDGES * HEADS, B), B, 0, stream>>>(src, dst, scores, ebuf, mbuf);
    k_edge_exp<<<G((long)N_EDGES * HEADS, B), B, 0, stream>>>(dst, mbuf, ebuf, zbuf);
    k_edge_msg<<<G((long)N_EDGES * EMB, B), B, 0, stream>>>(src, dst, rel, ebuf, zbuf,
                                                            feat, relmsg, agg);
    gemm64_wmma<<<gemm_blocks, 32 * GEMM_WAVES, 0, stream>>>(
        hin, loop_w + (size_t)l * EMB * EMB, loopA, N_NODES, EMB);
    gemm64_wmma<<<gemm_blocks, 32 * GEMM_WAVES, 0, stream>>>(
        hin, evol_w + (size_t)l * EMB * EMB, loopB, N_NODES, EMB);
    k_combine<<<G((long)N_NODES * EMB, B), B, 0, stream>>>(agg, loopA, loopB, indeg, hout);
    float* t = hin; hin = hout; hout = t;
  }

  k_zero_f32<<<G(NUM_GRAPHS * EMB, B), B, 0, stream>>>(out, NUM_GRAPHS * EMB);
  k_readout<<<G((long)N_NODES * EMB, B), B, 0, stream>>>(graph_id, hin, rd_w, out);
}